// ORCDF_43224550867566
// MI455X (gfx1250) — compile-verified
//
#include <hip/hip_runtime.h>
#include <cstddef>

typedef __attribute__((ext_vector_type(2))) float v2f;
typedef __attribute__((ext_vector_type(8))) float v8f;

#define NS 8000
#define NE 6000
#define NK 256
#define ND 64
#define NTOT (NS + NE + NK)   // 14256
#define NB 4096
#define NH1 512
#define NH2 256
#define SSLW 0.01f
#define INVTEMP 1.25f         // 1/0.8

// ---------------- elementwise helpers ----------------

__global__ void orcdf_zero(float* p, int n) {
  int i = blockIdx.x * blockDim.x + threadIdx.x;
  if (i < n) p[i] = 0.0f;
}

__global__ void orcdf_scale(float* p, int n, float s) {
  int i = blockIdx.x * blockDim.x + threadIdx.x;
  if (i < n) p[i] *= s;
}

// build emb0 = concat(stu,exer,know) into r, w, acc
__global__ void orcdf_concat_init(const float* __restrict__ stu,
                                  const float* __restrict__ exer,
                                  const float* __restrict__ know,
                                  float* __restrict__ r, float* __restrict__ w,
                                  float* __restrict__ acc, int total) {
  int i = blockIdx.x * blockDim.x + threadIdx.x;
  if (i >= total) return;
  int n = i >> 6, d = i & 63;
  float v;
  if (n < NS)            v = stu[i];
  else if (n < NS + NE)  v = exer[(n - NS) * ND + d];
  else                   v = know[(n - NS - NE) * ND + d];
  r[i] = v; w[i] = v; acc[i] = v;
}

// X[n,0:64]=r[n], X[n,64:128]=w[n]
__global__ void orcdf_concat2(const float* __restrict__ r, const float* __restrict__ w,
                              float* __restrict__ X, int total) {
  int i = blockIdx.x * blockDim.x + threadIdx.x;
  if (i >= total) return;
  int n = i >> 6, d = i & 63;
  X[(size_t)n * 128 + d]      = r[i];
  X[(size_t)n * 128 + 64 + d] = w[i];
}

// out[b,d] = emb[(rowoff+ids[b])*64+d] * wv[d]
__global__ void orcdf_gatherA(const float* __restrict__ emb, int rowoff,
                              const int* __restrict__ ids, const float* __restrict__ wv,
                              float* __restrict__ out, int total) {
  int i = blockIdx.x * blockDim.x + threadIdx.x;
  if (i >= total) return;
  int b = i >> 6, d = i & 63;
  out[i] = emb[(size_t)(rowoff + ids[b]) * ND + d] * wv[d];
}

// input_x = sigmoid(disc[ex[b]]) * (stat - kdiff) * q   (written into stat)
__global__ void orcdf_inputx(float* __restrict__ stat, const float* __restrict__ kdif,
                             const float* __restrict__ disc, const int* __restrict__ exid,
                             const float* __restrict__ q, int total) {
  int i = blockIdx.x * blockDim.x + threadIdx.x;
  if (i >= total) return;
  int b = i >> 8;  // K = 256
  float dsc = 1.0f / (1.0f + __expf(-disc[exid[b]]));
  stat[i] = dsc * (stat[i] - kdif[i]) * q[i];
}

// ---------------- SpMM: y[rows[e]] += vals[e] * x[cols[e]] ----------------

__global__ void orcdf_spmm(const int* __restrict__ rows, const int* __restrict__ cols,
                           const float* __restrict__ vals, const float* __restrict__ x,
                           float* __restrict__ y, int nnz) {
  int t = blockIdx.x * blockDim.x + threadIdx.x;
  int e = t >> 2;
  if (e >= nnz) return;
  int c16 = (t & 3) * 16;
  int r = rows[e], c = cols[e];
  float v = vals[e];
  const float* xs = x + (size_t)c * ND + c16;
  float* yd = y + (size_t)r * ND + c16;
#pragma unroll
  for (int i = 0; i < 16; i += 4) {
    float4 xv = *(const float4*)(xs + i);
    atomicAdd(yd + i + 0, v * xv.x);
    atomicAdd(yd + i + 1, v * xv.y);
    atomicAdd(yd + i + 2, v * xv.z);
    atomicAdd(yd + i + 3, v * xv.w);
  }
}

// ---------------- generic WMMA f32 GEMM: C = act(A[M,K] @ W.T + bias) ----------------
// W row-major [Nn, Kk]  =>  B[k][n] = W[n*Kk + k].  One wave per 16x16 tile.
// act: 0=none, 1=tanh, 2=sigmoid.  accumulate: C += x instead of C = x.

__global__ void orcdf_gemm_wt(const float* __restrict__ A, const float* __restrict__ W,
                              const float* __restrict__ bias, int bias_stride,
                              float* __restrict__ C, int Kk, int Nn,
                              int act, int accumulate) {
  const int lane = threadIdx.x;
  const int l15 = lane & 15, half = lane >> 4;
  const int row0 = blockIdx.x * 16, col0 = blockIdx.y * 16;
  const float* arow = A + (size_t)(row0 + l15) * Kk + half * 2;
  const float* brow = W + (size_t)(col0 + l15) * Kk + half * 2;
  v8f c = {};
  for (int k0 = 0; k0 < Kk; k0 += 4) {
    v2f a, b;
    a.x = arow[k0]; a.y = arow[k0 + 1];
    b.x = brow[k0]; b.y = brow[k0 + 1];
    c = __builtin_amdgcn_wmma_f32_16x16x4_f32(false, a, false, b, (short)0, c, false, false);
  }
#pragma unroll
  for (int v = 0; v < 8; ++v) {
    int mrow = row0 + half * 8 + v;
    int ncol = col0 + l15;
    float x = c[v];
    x += bias[ncol * bias_stride];
    if (act == 1)      x = tanhf(x);
    else if (act == 2) x = 1.0f / (1.0f + __expf(-x));
    size_t idx = (size_t)mrow * Nn + ncol;
    if (accumulate) C[idx] += x; else C[idx] = x;
  }
}

// ---------------- streaming InfoNCE (online softmax over WMMA logit tiles) ----------------

__device__ inline float half16_max(float x) {
  x = fmaxf(x, __shfl_xor(x, 1, 16));
  x = fmaxf(x, __shfl_xor(x, 2, 16));
  x = fmaxf(x, __shfl_xor(x, 4, 16));
  x = fmaxf(x, __shfl_xor(x, 8, 16));
  return x;
}
__device__ inline float half16_sum(float x) {
  x += __shfl_xor(x, 1, 16);
  x += __shfl_xor(x, 2, 16);
  x += __shfl_xor(x, 4, 16);
  x += __shfl_xor(x, 8, 16);
  return x;
}
__device__ inline float wave32_sum(float x) {
  x += __shfl_xor(x, 1, 32);
  x += __shfl_xor(x, 2, 32);
  x += __shfl_xor(x, 4, 32);
  x += __shfl_xor(x, 8, 32);
  x += __shfl_xor(x, 16, 32);
  return x;
}

// sum over rows of -(log_softmax(v1 @ v2.T / T))[i,i]; one wave handles 16 rows
__global__ void orcdf_nce(const float* __restrict__ v1, const float* __restrict__ v2,
                          int n, float* __restrict__ lossAcc, int which) {
  const int lane = threadIdx.x;
  const int l15 = lane & 15, half = lane >> 4;
  const int row0 = blockIdx.x * 16;

  // preload A fragments for all 16 k-steps (K = 64)
  v2f af[16];
  const float* arow = v1 + (size_t)(row0 + l15) * ND + half * 2;
#pragma unroll
  for (int kk = 0; kk < 16; ++kk) {
    af[kk].x = arow[kk * 4];
    af[kk].y = arow[kk * 4 + 1];
  }

  float m[8], s[8], dg[8];
#pragma unroll
  for (int v = 0; v < 8; ++v) { m[v] = -3.0e38f; s[v] = 0.0f; dg[v] = 0.0f; }

  for (int col0 = 0; col0 < n; col0 += 16) {
    v8f c = {};
    const float* brow = v2 + (size_t)(col0 + l15) * ND + half * 2;
#pragma unroll
    for (int kk = 0; kk < 16; ++kk) {
      v2f b;
      b.x = brow[kk * 4];
      b.y = brow[kk * 4 + 1];
      c = __builtin_amdgcn_wmma_f32_16x16x4_f32(false, af[kk], false, b, (short)0, c, false, false);
    }
#pragma unroll
    for (int v = 0; v < 8; ++v) {
      float x = c[v] * INVTEMP;
      int mrow = half * 8 + v;
      if (col0 + l15 == row0 + mrow) dg[v] += x;   // diagonal element of the big matrix
      float tmax = half16_max(x);
      float newm = fmaxf(m[v], tmax);
      float psum = half16_sum(__expf(x - newm));
      s[v] = s[v] * __expf(m[v] - newm) + psum;
      m[v] = newm;
    }
  }

  float t = 0.0f;
#pragma unroll
  for (int v = 0; v < 8; ++v) {
    float d = half16_sum(dg[v]);                    // broadcast the single nonzero lane
    t += m[v] + __logf(s[v]) - d;                   // -(diag - logsumexp)
  }
  float contrib = (l15 == 0) ? t : 0.0f;            // one lane per half contributes
  contrib = wave32_sum(contrib);
  if (lane == 0) atomicAdd(&lossAcc[which], contrib);
}

// ---------------- tiny tails ----------------

__global__ void orcdf_pred(const float* __restrict__ h, const float* __restrict__ W2,
                           const float* __restrict__ b2, float* __restrict__ out) {
  int b = blockIdx.x * blockDim.x + threadIdx.x;
  if (b >= NB) return;
  float sacc = b2[0];
  const float* hr = h + (size_t)b * NH2;
  for (int k = 0; k < NH2; ++k) sacc += hr[k] * W2[k];
  out[b] = 1.0f / (1.0f + __expf(-sacc));
}

__global__ void orcdf_finloss(const float* __restrict__ lossAcc, float* __restrict__ out) {
  out[0] = SSLW * (lossAcc[0] * (1.0f / NS) + lossAcc[1] * (1.0f / NE));
}

// ---------------- host launch ----------------

extern "C" void kernel_launch(void* const* d_in, const int* in_sizes, int n_in,
                              void* d_out, int out_size, void* d_ws, size_t ws_size,
                              hipStream_t stream) {
  const int* rR = (const int*)d_in[0];   const int* rC = (const int*)d_in[1];
  const float* rV = (const float*)d_in[2];
  const int* wR = (const int*)d_in[3];   const int* wC = (const int*)d_in[4];
  const float* wV = (const float*)d_in[5];
  const int* frR = (const int*)d_in[6];  const int* frC = (const int*)d_in[7];
  const float* frV = (const float*)d_in[8];
  const int* fwR = (const int*)d_in[9];  const int* fwC = (const int*)d_in[10];
  const float* fwV = (const float*)d_in[11];
  const int* stu_id = (const int*)d_in[12];
  const int* ex_id  = (const int*)d_in[13];
  const float* qmask = (const float*)d_in[14];
  const float* stu_emb = (const float*)d_in[15];
  const float* exer_emb = (const float*)d_in[16];
  const float* know_emb = (const float*)d_in[17];
  const float* disc_emb = (const float*)d_in[18];
  const float* Wc = (const float*)d_in[19];  const float* bc = (const float*)d_in[20];
  const float* w_stat = (const float*)d_in[21]; const float* b_stat = (const float*)d_in[22];
  const float* w_kdiff = (const float*)d_in[23]; const float* b_kdiff = (const float*)d_in[24];
  const float* W0 = (const float*)d_in[25]; const float* b0 = (const float*)d_in[26];
  const float* W1 = (const float*)d_in[27]; const float* b1 = (const float*)d_in[28];
  const float* W2 = (const float*)d_in[29]; const float* b2 = (const float*)d_in[30];

  const int nnz = in_sizes[0];
  const int N64 = NTOT * ND;            // 912384

  float* ws = (float*)d_ws;
  size_t o = 0;
  float* r_cur = ws + o; o += N64;
  float* w_cur = ws + o; o += N64;
  float* r_nxt = ws + o; o += N64;
  float* w_nxt = ws + o; o += N64;
  float* outF  = ws + o; o += N64;
  float* outFF = ws + o; o += N64;
  float* X     = ws + o; o += (size_t)NTOT * 128;
  float* statA = ws + o; o += (size_t)NB * ND;
  float* kdifA = ws + o; o += (size_t)NB * ND;
  float* statB = ws + o; o += (size_t)NB * NK;
  float* kdifB = ws + o; o += (size_t)NB * NK;
  float* h0    = ws + o; o += (size_t)NB * NH1;
  float* h1    = ws + o; o += (size_t)NB * NH2;
  float* lossAcc = ws + o; o += 8;

  const int EB = 256;
  const int gN64 = (N64 + EB - 1) / EB;
  const int gSp  = (nnz * 4 + EB - 1) / EB;

  // ---- one GCN forward pass ----
  auto forward = [&](const int* gr_r, const int* gr_c, const float* gr_v,
                     const int* gw_r, const int* gw_c, const float* gw_v,
                     float* acc) {
    orcdf_concat_init<<<gN64, EB, 0, stream>>>(stu_emb, exer_emb, know_emb,
                                               r_cur, w_cur, acc, N64);
    for (int l = 0; l < 3; ++l) {
      orcdf_zero<<<gN64, EB, 0, stream>>>(r_nxt, N64);
      orcdf_zero<<<gN64, EB, 0, stream>>>(w_nxt, N64);
      orcdf_spmm<<<gSp, EB, 0, stream>>>(gr_r, gr_c, gr_v, r_cur, r_nxt, nnz);
      orcdf_spmm<<<gSp, EB, 0, stream>>>(gw_r, gw_c, gw_v, w_cur, w_nxt, nnz);
      orcdf_concat2<<<gN64, EB, 0, stream>>>(r_nxt, w_nxt, X, N64);
      // acc += X @ Wc.T + bc
      dim3 gc(NTOT / 16, ND / 16);
      orcdf_gemm_wt<<<gc, 32, 0, stream>>>(X, Wc, bc, 1, acc, 128, ND, 0, 1);
      float* t;
      t = r_cur; r_cur = r_nxt; r_nxt = t;
      t = w_cur; w_cur = w_nxt; w_nxt = t;
    }
    orcdf_scale<<<gN64, EB, 0, stream>>>(acc, N64, 0.25f);  // mean of 4 embeddings
  };

  forward(rR, rC, rV, wR, wC, wV, outF);
  forward(frR, frC, frV, fwR, fwC, fwV, outFF);

  // ---- InfoNCE losses (streaming, WMMA logit tiles) ----
  orcdf_zero<<<1, 8, 0, stream>>>(lossAcc, 8);
  orcdf_nce<<<NS / 16, 32, 0, stream>>>(outF, outFF, NS, lossAcc, 0);
  orcdf_nce<<<NE / 16, 32, 0, stream>>>(outF + (size_t)NS * ND,
                                        outFF + (size_t)NS * ND, NE, lossAcc, 1);

  // ---- KANCD head ----
  const float* knowF = outF + (size_t)(NS + NE) * ND;     // [256, 64]
  const int gB64 = (NB * ND + EB - 1) / EB;
  orcdf_gatherA<<<gB64, EB, 0, stream>>>(outF, 0, stu_id, w_stat, statA, NB * ND);
  orcdf_gatherA<<<gB64, EB, 0, stream>>>(outF, NS, ex_id, w_kdiff, kdifA, NB * ND);

  dim3 gk(NB / 16, NK / 16);
  orcdf_gemm_wt<<<gk, 32, 0, stream>>>(statA, knowF, b_stat, 0, statB, ND, NK, 2, 0);
  orcdf_gemm_wt<<<gk, 32, 0, stream>>>(kdifA, knowF, b_kdiff, 0, kdifB, ND, NK, 2, 0);

  const int gBK = (NB * NK + EB - 1) / EB;
  orcdf_inputx<<<gBK, EB, 0, stream>>>(statB, kdifB, disc_emb, ex_id, qmask, NB * NK);

  dim3 g0(NB / 16, NH1 / 16);
  orcdf_gemm_wt<<<g0, 32, 0, stream>>>(statB, W0, b0, 1, h0, NK, NH1, 1, 0);
  dim3 g1(NB / 16, NH2 / 16);
  orcdf_gemm_wt<<<g1, 32, 0, stream>>>(h0, W1, b1, 1, h1, NH1, NH2, 1, 0);

  float* out = (float*)d_out;
  orcdf_pred<<<NB / EB, EB, 0, stream>>>(h1, W2, b2, out);
  orcdf_finloss<<<1, 1, 0, stream>>>(lossAcc, out + NB);
}